// GAT_11321533792501
// MI455X (gfx1250) — compile-verified
//
#include <hip/hip_runtime.h>
#include <math.h>

#define NN   50000
#define FIN  256
#define HID  256      // H*C
#define HH   4
#define CC   64
#define NC   40
#define SLOPE 0.2f

typedef __attribute__((ext_vector_type(2))) float v2f;
typedef __attribute__((ext_vector_type(8))) float v8f;

__device__ __forceinline__ float leaky(float v) { return v > 0.f ? v : SLOPE * v; }

__device__ __forceinline__ void atomicMaxF(float* a, float v) {
    int* ai = (int*)a;
    int cur = __float_as_int(*a);
    while (v > __int_as_float(cur)) {
        int prev = atomicCAS(ai, cur, __float_as_int(v));
        if (prev == cur) break;
        cur = prev;
    }
}

__device__ __forceinline__ void edge_sd(const int* __restrict__ EI, int e, int Eraw,
                                        int& s, int& d) {
    if (e < Eraw) { s = EI[e]; d = EI[Eraw + e]; }
    else          { s = e - Eraw; d = s; }
}

// ---------------- init kernels ----------------
__global__ void init1_k(float* __restrict__ Out1, const float* __restrict__ b1,
                        float* __restrict__ amax1, float* __restrict__ den1) {
    int i = blockIdx.x * blockDim.x + threadIdx.x;
    if (i < NN * HID) Out1[i] = b1[i & (HID - 1)];
    if (i < NN * HH) { amax1[i] = -INFINITY; den1[i] = 0.f; }
}

__global__ void init2_k(float* __restrict__ OutF, const float* __restrict__ b2,
                        float* __restrict__ amax2, float* __restrict__ den2) {
    int i = blockIdx.x * blockDim.x + threadIdx.x;
    if (i < NN * NC) OutF[i] = b2[i % NC];
    if (i < NN) { amax2[i] = -INFINITY; den2[i] = 0.f; }
}

// ---------------- GEMM1: h1[N,256] = X[N,256] @ W1[256,256] ----------------
// block = 128 threads (4 waves); block handles 16 rows x 256 cols; wave w -> cols [64w,64w+64)
__global__ __launch_bounds__(128) void gemm1_wmma(const float* __restrict__ X,
                                                  const float* __restrict__ W,
                                                  float* __restrict__ Hout) {
    const int lane = threadIdx.x & 31;
    const int wave = threadIdx.x >> 5;
    const int m    = lane & 15;
    const int hi   = lane >> 4;
    const int row0 = blockIdx.x << 4;

    v8f acc[4] = {};
    const float* arow = X + (size_t)(row0 + m) * FIN + 2 * hi;
    for (int k0 = 0; k0 < FIN; k0 += 4) {
        v2f a;
        a.x = arow[k0];
        a.y = arow[k0 + 1];
        const int kb = k0 + 2 * hi;
        const float* bp = W + (size_t)kb * HID + wave * 64 + m;
#pragma unroll
        for (int t = 0; t < 4; t++) {
            v2f b;
            b.x = bp[t * 16];
            b.y = bp[HID + t * 16];
            acc[t] = __builtin_amdgcn_wmma_f32_16x16x4_f32(
                false, a, false, b, (short)0, acc[t], false, false);
        }
    }
#pragma unroll
    for (int t = 0; t < 4; t++)
#pragma unroll
        for (int v = 0; v < 8; v++)
            Hout[(size_t)(row0 + v + 8 * hi) * HID + wave * 64 + t * 16 + m] = acc[t][v];
}

// ---------------- GEMM2: g2[N,40] = X2[N,256] @ W2[256,40] ----------------
// block = 32 threads (1 wave); 16 rows x 48 cols (3 tiles, guarded at col 40)
__global__ __launch_bounds__(32) void gemm2_wmma(const float* __restrict__ X2,
                                                 const float* __restrict__ W2,
                                                 float* __restrict__ G) {
    const int lane = threadIdx.x & 31;
    const int m    = lane & 15;
    const int hi   = lane >> 4;
    const int row0 = blockIdx.x << 4;

    v8f acc[3] = {};
    const float* arow = X2 + (size_t)(row0 + m) * HID + 2 * hi;
    for (int k0 = 0; k0 < HID; k0 += 4) {
        v2f a;
        a.x = arow[k0];
        a.y = arow[k0 + 1];
        const int kb = k0 + 2 * hi;
#pragma unroll
        for (int t = 0; t < 3; t++) {
            const int n = t * 16 + m;
            v2f b;
            b.x = (n < NC) ? W2[(size_t)kb * NC + n] : 0.f;
            b.y = (n < NC) ? W2[(size_t)(kb + 1) * NC + n] : 0.f;
            acc[t] = __builtin_amdgcn_wmma_f32_16x16x4_f32(
                false, a, false, b, (short)0, acc[t], false, false);
        }
    }
#pragma unroll
    for (int t = 0; t < 3; t++) {
        const int n = t * 16 + m;
        if (n < NC)
#pragma unroll
            for (int v = 0; v < 8; v++)
                G[(size_t)(row0 + v + 8 * hi) * NC + n] = acc[t][v];
    }
}

// ---------------- attention scalar dots ----------------
__global__ void att1_k(const float* __restrict__ H1, const float* __restrict__ atts,
                       const float* __restrict__ attd,
                       float* __restrict__ AS, float* __restrict__ AD) {
    int i = blockIdx.x * blockDim.x + threadIdx.x;
    if (i >= NN * HH) return;
    const int n = i >> 2, h = i & 3;
    const float* hp = H1 + (size_t)n * HID + h * CC;
    const float* sp = atts + h * CC;
    const float* dp = attd + h * CC;
    float s = 0.f, d = 0.f;
    for (int c = 0; c < CC; c++) { s += hp[c] * sp[c]; d += hp[c] * dp[c]; }
    AS[i] = s; AD[i] = d;
}

__global__ void att2_k(const float* __restrict__ G, const float* __restrict__ atts,
                       const float* __restrict__ attd,
                       float* __restrict__ AS, float* __restrict__ AD) {
    int n = blockIdx.x * blockDim.x + threadIdx.x;
    if (n >= NN) return;
    const float* gp = G + (size_t)n * NC;
    float s = 0.f, d = 0.f;
    for (int c = 0; c < NC; c++) { s += gp[c] * atts[c]; d += gp[c] * attd[c]; }
    AS[n] = s; AD[n] = d;
}

// ---------------- edge passes, layer 1 (H=4) ----------------
__global__ void edge_max1(const int* __restrict__ EI, int Eraw, int Etot,
                          const float* __restrict__ AS, const float* __restrict__ AD,
                          float* __restrict__ AMAX) {
    int e = blockIdx.x * blockDim.x + threadIdx.x;
    if (e >= Etot) return;
    int s, d; edge_sd(EI, e, Eraw, s, d);
#pragma unroll
    for (int h = 0; h < HH; h++) {
        float a = leaky(AS[s * HH + h] + AD[d * HH + h]);
        atomicMaxF(&AMAX[d * HH + h], a);
    }
}

__global__ void edge_sum1(const int* __restrict__ EI, int Eraw, int Etot,
                          const float* __restrict__ AS, const float* __restrict__ AD,
                          const float* __restrict__ AMAX, float* __restrict__ DEN) {
    int e = blockIdx.x * blockDim.x + threadIdx.x;
    if (e >= Etot) return;
    int s, d; edge_sd(EI, e, Eraw, s, d);
#pragma unroll
    for (int h = 0; h < HH; h++) {
        float a = leaky(AS[s * HH + h] + AD[d * HH + h]);
        atomicAdd(&DEN[d * HH + h], expf(a - AMAX[d * HH + h]));
    }
}

// wave per edge: 256 channels, lane-strided float atomics
__global__ __launch_bounds__(256) void edge_aggr1(
        const int* __restrict__ EI, int Eraw, int Etot,
        const float* __restrict__ H1,
        const float* __restrict__ AS, const float* __restrict__ AD,
        const float* __restrict__ AMAX, const float* __restrict__ DEN,
        float* __restrict__ Out) {
    int wid  = blockIdx.x * (blockDim.x >> 5) + (threadIdx.x >> 5);
    int lane = threadIdx.x & 31;
    if (wid >= Etot) return;
    int s, d; edge_sd(EI, wid, Eraw, s, d);
    float w[HH];
#pragma unroll
    for (int h = 0; h < HH; h++) {
        float a = leaky(AS[s * HH + h] + AD[d * HH + h]);
        w[h] = expf(a - AMAX[d * HH + h]) / (DEN[d * HH + h] + 1e-16f);
    }
    const float* hs = H1 + (size_t)s * HID;
    float* od = Out + (size_t)d * HID;
#pragma unroll
    for (int c0 = 0; c0 < HID; c0 += 32) {
        int c = c0 + lane;
        atomicAdd(&od[c], hs[c] * w[c >> 6]);
    }
}

__global__ void elu_k(float* __restrict__ X) {
    int i = blockIdx.x * blockDim.x + threadIdx.x;
    if (i >= NN * HID) return;
    float v = X[i];
    X[i] = v > 0.f ? v : (expf(v) - 1.f);
}

// ---------------- edge passes, layer 2 (H=1, C=40) ----------------
__global__ void edge_max2(const int* __restrict__ EI, int Eraw, int Etot,
                          const float* __restrict__ AS, const float* __restrict__ AD,
                          float* __restrict__ AMAX) {
    int e = blockIdx.x * blockDim.x + threadIdx.x;
    if (e >= Etot) return;
    int s, d; edge_sd(EI, e, Eraw, s, d);
    atomicMaxF(&AMAX[d], leaky(AS[s] + AD[d]));
}

__global__ void edge_sum2(const int* __restrict__ EI, int Eraw, int Etot,
                          const float* __restrict__ AS, const float* __restrict__ AD,
                          const float* __restrict__ AMAX, float* __restrict__ DEN) {
    int e = blockIdx.x * blockDim.x + threadIdx.x;
    if (e >= Etot) return;
    int s, d; edge_sd(EI, e, Eraw, s, d);
    atomicAdd(&DEN[d], expf(leaky(AS[s] + AD[d]) - AMAX[d]));
}

__global__ __launch_bounds__(256) void edge_aggr2(
        const int* __restrict__ EI, int Eraw, int Etot,
        const float* __restrict__ G,
        const float* __restrict__ AS, const float* __restrict__ AD,
        const float* __restrict__ AMAX, const float* __restrict__ DEN,
        float* __restrict__ Out) {
    int wid  = blockIdx.x * (blockDim.x >> 5) + (threadIdx.x >> 5);
    int lane = threadIdx.x & 31;
    if (wid >= Etot) return;
    int s, d; edge_sd(EI, wid, Eraw, s, d);
    float w = expf(leaky(AS[s] + AD[d]) - AMAX[d]) / (DEN[d] + 1e-16f);
    const float* gs = G + (size_t)s * NC;
    float* od = Out + (size_t)d * NC;
    for (int c = lane; c < NC; c += 32) atomicAdd(&od[c], gs[c] * w);
}

// ---------------- launch ----------------
extern "C" void kernel_launch(void* const* d_in, const int* in_sizes, int n_in,
                              void* d_out, int out_size, void* d_ws, size_t ws_size,
                              hipStream_t stream) {
    const float* X     = (const float*)d_in[0];
    const int*   EI    = (const int*)d_in[1];
    const float* W1    = (const float*)d_in[2];
    const float* atts1 = (const float*)d_in[3];
    const float* attd1 = (const float*)d_in[4];
    const float* b1    = (const float*)d_in[5];
    const float* W2    = (const float*)d_in[6];
    const float* atts2 = (const float*)d_in[7];
    const float* attd2 = (const float*)d_in[8];
    const float* b2    = (const float*)d_in[9];

    const int Eraw = in_sizes[1] / 2;
    const int Etot = Eraw + NN;

    float* ws = (float*)d_ws;
    // layout (floats); layer-2 buffers alias layer-1 ones after they are dead
    float* h1    = ws;                         // N*256  (reused as g2 in layer 2)
    float* out1  = h1 + (size_t)NN * HID;      // N*256
    float* asrc1 = out1 + (size_t)NN * HID;    // N*4  (reused as asrc2: N)
    float* adst1 = asrc1 + (size_t)NN * HH;    // N*4  (reused as adst2: N)
    float* amax1 = adst1 + (size_t)NN * HH;    // N*4  (reused as amax2: N)
    float* den1  = amax1 + (size_t)NN * HH;    // N*4  (reused as den2:  N)
    float* g2    = h1;
    float* asrc2 = asrc1;
    float* adst2 = adst1;
    float* amax2 = amax1;
    float* den2  = den1;
    float* outF  = (float*)d_out;

    const int TB = 256;
    const int gridNH  = (NN * HID + TB - 1) / TB;
    const int gridE   = (Etot + TB - 1) / TB;
    const int gridEw  = (Etot + 7) / 8;          // 8 waves per 256-thread block
    const int gridRow = NN / 16;                 // 50000 = 16 * 3125

    // ---- layer 1 ----
    init1_k<<<gridNH, TB, 0, stream>>>(out1, b1, amax1, den1);
    gemm1_wmma<<<gridRow, 128, 0, stream>>>(X, W1, h1);
    att1_k<<<(NN * HH + TB - 1) / TB, TB, 0, stream>>>(h1, atts1, attd1, asrc1, adst1);
    edge_max1<<<gridE, TB, 0, stream>>>(EI, Eraw, Etot, asrc1, adst1, amax1);
    edge_sum1<<<gridE, TB, 0, stream>>>(EI, Eraw, Etot, asrc1, adst1, amax1, den1);
    edge_aggr1<<<gridEw, TB, 0, stream>>>(EI, Eraw, Etot, h1, asrc1, adst1, amax1, den1, out1);
    elu_k<<<gridNH, TB, 0, stream>>>(out1);

    // ---- layer 2 ----
    init2_k<<<(NN * NC + TB - 1) / TB, TB, 0, stream>>>(outF, b2, amax2, den2);
    gemm2_wmma<<<gridRow, 32, 0, stream>>>(out1, W2, g2);
    att2_k<<<(NN + TB - 1) / TB, TB, 0, stream>>>(g2, atts2, attd2, asrc2, adst2);
    edge_max2<<<gridE, TB, 0, stream>>>(EI, Eraw, Etot, asrc2, adst2, amax2);
    edge_sum2<<<gridE, TB, 0, stream>>>(EI, Eraw, Etot, asrc2, adst2, amax2, den2);
    edge_aggr2<<<gridEw, TB, 0, stream>>>(EI, Eraw, Etot, g2, asrc2, adst2, amax2, den2, outF);
}